// MyMessagePassing_7069516169579
// MI455X (gfx1250) — compile-verified
//
#include <hip/hip_runtime.h>
#include <stdint.h>

#define D_FEAT 128
#define EDGES_PER_BLOCK 256

// ---------------------------------------------------------------------------
// Kernel 1: zero the output (harness poisons d_out with 0xAA).
// ---------------------------------------------------------------------------
__global__ void gnn_zero_kernel(float* __restrict__ out, int n) {
    int i = (blockIdx.x * blockDim.x + threadIdx.x) * 4;
    if (i + 3 < n) {
        float4 z = make_float4(0.f, 0.f, 0.f, 0.f);
        *reinterpret_cast<float4*>(out + i) = z;
    } else {
        for (; i < n; ++i) out[i] = 0.f;
    }
}

// ---------------------------------------------------------------------------
// Kernel 2: edge-parallel gather + scatter-add.
//   block = 256 threads = 8 waves (wave32); block covers 256 edges.
//   Each wave stages its own 32 edges' (src,dst) into LDS with CDNA5
//   global_load_async_to_lds_b32 (ASYNCcnt) + s_wait_asynccnt, then
//   processes one edge per step: lane l moves floats [4l, 4l+4) of the row
//   with one global_load_b128 and four native global_atomic_add_f32.
// ---------------------------------------------------------------------------
__global__ __launch_bounds__(256) void gnn_scatter_kernel(
    const float* __restrict__ x,
    const int*   __restrict__ edge_index,   // [2, E] : row 0 = src, row 1 = dst
    float*       __restrict__ out,
    int n_edges) {

    __shared__ int smem[2 * EDGES_PER_BLOCK];   // [0,256): src   [256,512): dst

    const int t     = threadIdx.x;
    const int lane  = t & 31;
    const int wave  = t >> 5;
    const int eBase = blockIdx.x * EDGES_PER_BLOCK;
    const int e     = eBase + t;

    // ---- stage edge indices: per-lane async global -> LDS copy (CDNA5) ----
    if (e < n_edges) {
        uint32_t lds_src = (uint32_t)(uintptr_t)(&smem[t]);
        uint32_t lds_dst = (uint32_t)(uintptr_t)(&smem[EDGES_PER_BLOCK + t]);
        uint64_t g_src   = (uint64_t)(uintptr_t)(edge_index + e);
        uint64_t g_dst   = (uint64_t)(uintptr_t)(edge_index + n_edges + e);
        asm volatile(
            "global_load_async_to_lds_b32 %0, %2, off\n\t"
            "global_load_async_to_lds_b32 %1, %3, off\n\t"
            "s_wait_asynccnt 0x0\n\t"
            :
            : "v"(lds_src), "v"(lds_dst), "v"(g_src), "v"(g_dst)
            : "memory");
    }
    // Each wave reads back only the LDS words its own lanes staged, so its
    // own ASYNCcnt wait is sufficient -- no workgroup barrier required.

    const float4* __restrict__ x4 = reinterpret_cast<const float4*>(x);
    const int base = wave * 32;          // this wave's 32 edges within block

    #pragma unroll 8
    for (int j = 0; j < 32; ++j) {
        if (eBase + base + j >= n_edges) break;
        // uniform-address LDS reads -> single-bank broadcast per wave
        const int s = smem[base + j];
        const int d = smem[EDGES_PER_BLOCK + base + j];

        // lane l loads floats [4l, 4l+4) of row s : one global_load_b128
        float4 v = x4[(size_t)s * (D_FEAT / 4) + lane];

        // four native no-return fp32 L2 atomics sharing one address pair.
        // No memory clobber: inputs are registers, so the scheduler may
        // hoist later iterations' loads above these for ILP.
        uint64_t oaddr = (uint64_t)(uintptr_t)(out + (size_t)d * D_FEAT + lane * 4);
        asm volatile(
            "global_atomic_add_f32 %0, %1, off\n\t"
            "global_atomic_add_f32 %0, %2, off offset:4\n\t"
            "global_atomic_add_f32 %0, %3, off offset:8\n\t"
            "global_atomic_add_f32 %0, %4, off offset:12\n\t"
            :
            : "v"(oaddr), "v"(v.x), "v"(v.y), "v"(v.z), "v"(v.w));
    }
}

// ---------------------------------------------------------------------------
extern "C" void kernel_launch(void* const* d_in, const int* in_sizes, int n_in,
                              void* d_out, int out_size, void* d_ws, size_t ws_size,
                              hipStream_t stream) {
    const float* x          = (const float*)d_in[0];
    const int*   edge_index = (const int*)d_in[1];
    float*       out        = (float*)d_out;

    const int n_edges = in_sizes[1] / 2;        // edge_index is [2, E]

    // 1) zero output (stream-ordered before the scatter kernel)
    {
        int n_vec4   = (out_size + 3) / 4;
        int threads  = 256;
        int blocks   = (n_vec4 + threads - 1) / threads;
        gnn_zero_kernel<<<blocks, threads, 0, stream>>>(out, out_size);
    }

    // 2) edge-parallel gather/scatter-add
    {
        int blocks = (n_edges + EDGES_PER_BLOCK - 1) / EDGES_PER_BLOCK;
        gnn_scatter_kernel<<<blocks, 256, 0, stream>>>(x, edge_index, out, n_edges);
    }
}